// HFRB_56186762166438
// MI455X (gfx1250) — compile-verified
//
#include <hip/hip_runtime.h>
#include <hip/hip_bf16.h>

// ---------------------------------------------------------------------------
// HFRB forward for MI455X (gfx1250, wave32, WMMA).
// Kernel 1: down-projection GEMM via V_WMMA_F32_16X16X4_F32, writes xd in
//           planar (B, Cd, H, W) layout to workspace (coalesced via LDS).
// Kernel 2: fused horizontal+vertical wavelet filtering + q2c quadrant
//           combine, writing the final (B, C, H, W) output directly (all
//           trailing reshapes in the reference are contiguity-preserving).
// ---------------------------------------------------------------------------

typedef float v2f __attribute__((ext_vector_type(2)));
typedef float v8f __attribute__((ext_vector_type(8)));

#define BIMG 4
#define HDIM 192
#define WDIM 192
#define CDIM 192
#define CD   128           // 2*C/3
#define KDIM 192           // = C
#define HW   (HDIM * WDIM) // 36864
#define H2   (HDIM / 2)    // 96
#define W2   (WDIM / 2)    // 96

__device__ __constant__ float H0F[13] = {
    -0.0017578125f, 0.0f, 0.022265625f, -0.046875f, -0.0482421875f,
    0.296875f, 0.55546875f, 0.296875f, -0.0482421875f, -0.046875f,
    0.022265625f, 0.0f, -0.0017578125f};

__device__ __constant__ float H1F[19] = {
    -7.06e-05f, 0.0f, 0.0013419f, -0.0018834f, -0.0071568f, 0.023856f,
    0.0556431f, -0.0516881f, -0.2997576f, 0.5594308f, -0.2997576f,
    -0.0516881f, 0.0556431f, 0.023856f, -0.0071568f, -0.0018834f,
    0.0013419f, 0.0f, -7.06e-05f};

// ---------------------------------------------------------------------------
// Kernel 1: xd = x @ w_down^T + b_down, stored planar (B, Cd, H, W).
// Block = 256 threads = 8 waves. Block owns 16 rows of M = B*H*W;
// wave i owns output columns [16i, 16i+16) of Cd = 128.
// A fragment (16x4 f32): lanes 0-15 hold M=0..15 with K=k,k+1;
//                        lanes 16-31 hold M=0..15 with K=k+2,k+3.
// B fragment (4x16 f32): mirrored with N instead of M (w_down rows are
// contiguous in K, so these are plain float2 loads).
// ---------------------------------------------------------------------------
__global__ __launch_bounds__(256) void hfrb_gemm_wmma(
    const float* __restrict__ x,      // (B*HW, K)
    const float* __restrict__ wdn,    // (Cd, K)
    const float* __restrict__ bdn,    // (Cd)
    float* __restrict__ xd)           // (B, Cd, H, W) planar
{
    const int wave = threadIdx.x >> 5;
    const int lane = threadIdx.x & 31;
    const int l15  = lane & 15;
    const int khalf = (lane >> 4) << 1;    // 0 or 2

    const int m0 = blockIdx.x * 16;        // 16 rows of M per block
    const int n0 = wave * 16;              // 16 cols of Cd per wave

    const float* arow = x   + (size_t)(m0 + l15) * KDIM + khalf;
    const float* brow = wdn + (size_t)(n0 + l15) * KDIM + khalf;

    v8f acc = {};
#pragma unroll
    for (int k = 0; k < KDIM; k += 4) {
        v2f a = *(const v2f*)(arow + k);
        v2f b = *(const v2f*)(brow + k);
        acc = __builtin_amdgcn_wmma_f32_16x16x4_f32(
            /*neg_a=*/false, a, /*neg_b=*/false, b,
            /*c_mod=*/(short)0, acc, /*reuse_a=*/false, /*reuse_b=*/false);
    }

    // D layout: lanes 0-15, VGPR r -> (M=r, N=lane); lanes 16-31 -> (M=r+8, N=lane-16)
    const float bv = bdn[n0 + l15];
    __shared__ float tile[16][CD + 4];     // 16 M-rows x 128 cd
    const int mrow0 = (lane >> 4) * 8;
#pragma unroll
    for (int r = 0; r < 8; ++r)
        tile[mrow0 + r][n0 + l15] = acc[r] + bv;
    __syncthreads();

    // Planar store: for fixed cd, 16 consecutive pixels are contiguous (64B runs).
    const int b_img = m0 / HW;             // HW % 16 == 0 -> tile never straddles images
    const int npix0 = m0 - b_img * HW;
    float* outbase = xd + (size_t)b_img * CD * HW + npix0;
    for (int idx = threadIdx.x; idx < 16 * CD; idx += 256) {
        const int cd = idx >> 4;           // threads 0..15 share cd -> contiguous stores
        const int r  = idx & 15;
        outbase[(size_t)cd * HW + r] = tile[r][cd];
    }
}

// ---------------------------------------------------------------------------
// Kernel 2: fused dual-tree wavelet stage.
// Grid: (w2/24, h2/24, B*Cd). Per workgroup: 24x24 output tile in (i,j)
// space -> 48x48 region of (h,w) -> 66x66 xd tile incl. +/-9 halo with
// symmetric (edge-included reflect) padding. Horizontal filtering commutes
// with the row reflection, so loading xd sym-padded in h is equivalent to
// sym-padding lo/hi before the vertical pass.
// LDS: 66*68 + 2*66*50 floats = 44,352 B.
// ---------------------------------------------------------------------------
__global__ __launch_bounds__(256) void hfrb_dtcwt(
    const float* __restrict__ xd,     // (B, Cd, H, W) planar
    float* __restrict__ out)          // (B, C, H, W) == (B, Cd*6, h2, w2) flat
{
    __shared__ float xt[66][68];
    __shared__ float lo[66][50];
    __shared__ float hi[66][50];

    const int plane = blockIdx.z;          // 0 .. B*Cd-1
    const int b  = plane >> 7;             // / Cd
    const int cd = plane & (CD - 1);
    const int i0 = blockIdx.y * 24;
    const int j0 = blockIdx.x * 24;

    const float* src = xd + (size_t)plane * HW;
    const int hb = 2 * i0 - 9;
    const int wb = 2 * j0 - 9;

    // ---- load 66x66 xd tile with symmetric padding ----
    for (int idx = threadIdx.x; idx < 66 * 66; idx += 256) {
        const int r = idx / 66, c = idx % 66;
        int hh = hb + r;
        hh = (hh < 0) ? (-hh - 1) : ((hh >= HDIM) ? (2 * HDIM - 1 - hh) : hh);
        int ww = wb + c;
        ww = (ww < 0) ? (-ww - 1) : ((ww >= WDIM) ? (2 * WDIM - 1 - ww) : ww);
        xt[r][c] = src[hh * WDIM + ww];
    }
    __syncthreads();

    // ---- horizontal pass: lo (13-tap H0), hi (19-tap H1) over 66 rows x 48 cols ----
    for (int idx = threadIdx.x; idx < 66 * 48; idx += 256) {
        const int r = idx / 48, j = idx % 48;  // j -> w = 2*j0 + j (local col j+9)
        float aHi = 0.f, aLo = 0.f;
#pragma unroll
        for (int t = 0; t < 19; ++t) aHi += H1F[t] * xt[r][j + t];
#pragma unroll
        for (int t = 0; t < 13; ++t) aLo += H0F[t] * xt[r][j + 3 + t];
        lo[r][j] = aLo;
        hi[r][j] = aHi;
    }
    __syncthreads();

    // ---- vertical pass + q2c quadrant combine, direct final-layout writes ----
    const float s = 0.70710678118654752f;   // 1/sqrt(2)
    const size_t obase = (size_t)b * (CDIM * HW) + (size_t)cd * 6 * (H2 * W2);
    for (int idx = threadIdx.x; idx < 24 * 24; idx += 256) {
        const int i = idx / 24, j = idx % 24;
        const int je = 2 * j, jo = 2 * j + 1;
        // even-sample local row for h=2*(i0+i) is 2i+9; odd is 2i+10
        float lh_e = 0.f, lh_o = 0.f, hh_e = 0.f, hh_o = 0.f;
#pragma unroll
        for (int t = 0; t < 19; ++t) {
            const float h1 = H1F[t];
            lh_e += h1 * lo[2 * i + t][je];
            lh_o += h1 * lo[2 * i + 1 + t][jo];
            hh_e += h1 * hi[2 * i + t][je];
            hh_o += h1 * hi[2 * i + 1 + t][jo];
        }
        float hl_e = 0.f, hl_o = 0.f;
#pragma unroll
        for (int t = 0; t < 13; ++t) {
            const float h0 = H0F[t];
            hl_e += h0 * hi[2 * i + 3 + t][je];
            hl_o += h0 * hi[2 * i + 4 + t][jo];
        }
        const size_t pix = (size_t)(i0 + i) * W2 + (j0 + j);
        out[obase + 0 * (H2 * W2) + pix] = (lh_e - lh_o) * s;  // r15
        out[obase + 1 * (H2 * W2) + pix] = (hh_e - hh_o) * s;  // r45
        out[obase + 2 * (H2 * W2) + pix] = (hl_e - hl_o) * s;  // r75
        out[obase + 3 * (H2 * W2) + pix] = (hl_e + hl_o) * s;  // r105
        out[obase + 4 * (H2 * W2) + pix] = (hh_e + hh_o) * s;  // r135
        out[obase + 5 * (H2 * W2) + pix] = (lh_e + lh_o) * s;  // r165
    }
}

extern "C" void kernel_launch(void* const* d_in, const int* in_sizes, int n_in,
                              void* d_out, int out_size, void* d_ws, size_t ws_size,
                              hipStream_t stream) {
    (void)in_sizes; (void)n_in; (void)out_size; (void)ws_size;
    const float* x   = (const float*)d_in[0];   // (B, H*W, C) fp32
    const float* wdn = (const float*)d_in[1];   // (Cd, C)    fp32
    const float* bdn = (const float*)d_in[2];   // (Cd)       fp32
    float* out = (float*)d_out;
    float* xd  = (float*)d_ws;                  // needs B*Cd*H*W*4 = 75.5 MB

    // GEMM: M = B*H*W = 147456 rows, 16 rows per block.
    hfrb_gemm_wmma<<<(BIMG * HW) / 16, 256, 0, stream>>>(x, wdn, bdn, xd);

    // Fused wavelet: h2/24 = w2/24 = 4 tiles, B*Cd = 512 planes.
    hfrb_dtcwt<<<dim3(W2 / 24, H2 / 24, BIMG * CD), 256, 0, stream>>>(xd, out);
}